// GAT_4501125726315
// MI455X (gfx1250) — compile-verified
//
#include <hip/hip_runtime.h>

#define N_NODES 50000
#define IN_CH   128
#define HC      128      // H*C
#define C_HID   64
#define N_EDGES 800000
#define OUT_CH  40
#define BN_EPS  1e-5f
#define NEG_SLOPE 0.2f

typedef float v2f __attribute__((ext_vector_type(2)));
typedef float v8f __attribute__((ext_vector_type(8)));

// ---------- monotonic float<->uint encoding for atomicMax on floats ----------
__device__ __forceinline__ unsigned f2mono(float f) {
    unsigned u = __float_as_uint(f);
    return (u & 0x80000000u) ? ~u : (u | 0x80000000u);
}
__device__ __forceinline__ float mono2f(unsigned u) {
    u = (u & 0x80000000u) ? (u & 0x7fffffffu) : ~u;
    return __uint_as_float(u);
}

__device__ __forceinline__ void edge_sd(const int* __restrict__ ei, int e, int E,
                                        int& s, int& d) {
    if (e < E) { s = ei[e]; d = ei[E + e]; }
    else       { s = d = e - E; }          // self loop
}

// ---------- BatchNorm statistics: one block per channel ----------
__global__ void bn_stats_kernel(const float* __restrict__ x, int n, int ncols,
                                float* __restrict__ mu, float* __restrict__ var) {
    int c = blockIdx.x;
    float s = 0.f, s2 = 0.f;
    for (int r = threadIdx.x; r < n; r += blockDim.x) {
        float v = x[(size_t)r * ncols + c];
        s += v; s2 += v * v;
    }
    __shared__ float sh[256], sh2[256];
    sh[threadIdx.x] = s; sh2[threadIdx.x] = s2;
    __syncthreads();
    for (int w = blockDim.x >> 1; w > 0; w >>= 1) {
        if (threadIdx.x < w) {
            sh[threadIdx.x]  += sh[threadIdx.x + w];
            sh2[threadIdx.x] += sh2[threadIdx.x + w];
        }
        __syncthreads();
    }
    if (threadIdx.x == 0) {
        float m = sh[0] / (float)n;
        mu[c]  = m;
        var[c] = sh2[0] / (float)n - m * m;   // biased variance
    }
}

// ---------- BatchNorm apply (+ optional ELU) ----------
__global__ void bn_apply_kernel(const float* __restrict__ x, float* __restrict__ y,
                                const float* __restrict__ mu, const float* __restrict__ var,
                                const float* __restrict__ g, const float* __restrict__ b,
                                long long total, int ncols, int do_elu) {
    long long i = (long long)blockIdx.x * blockDim.x + threadIdx.x;
    if (i >= total) return;
    int c = (int)(i % ncols);
    float v = (x[i] - mu[c]) * rsqrtf(var[c] + BN_EPS) * g[c] + b[c];
    if (do_elu) v = v > 0.f ? v : (__expf(v) - 1.f);
    y[i] = v;
}

// ---------- WMMA f32 GEMM: C[nrows,ncols] = A[nrows,K] @ B[K,ncols] ----------
// 128 threads = 4 waves; each wave computes one 16x16 tile with
// V_WMMA_F32_16X16X4_F32 over K in steps of 4. nrows%16==0, ncols%16==0.
__global__ void gemm_wmma_f32_kernel(const float* __restrict__ A,
                                     const float* __restrict__ B,
                                     float* __restrict__ Cm,
                                     int nrows, int K, int ncols) {
    int wave = threadIdx.x >> 5;
    int lane = threadIdx.x & 31;
    int row0 = blockIdx.x * 16;
    int col0 = (blockIdx.y * 4 + wave) * 16;
    if (col0 >= ncols || row0 >= nrows) return;   // wave-uniform

    int m    = lane & 15;        // row (A) / col (B,C) within tile
    int kb   = (lane >> 4) * 2;  // lanes 0-15 hold K 0..1, lanes 16-31 hold K 2..3
    int half = lane >> 4;

    v8f acc = {};
    const float* arow = A + (size_t)(row0 + m) * K;
    const float* bcol = B + col0 + m;
    for (int k0 = 0; k0 < K; k0 += 4) {
        v2f a, b;
        a.x = arow[k0 + kb];
        a.y = arow[k0 + kb + 1];
        b.x = bcol[(size_t)(k0 + kb) * ncols];
        b.y = bcol[(size_t)(k0 + kb + 1) * ncols];
        acc = __builtin_amdgcn_wmma_f32_16x16x4_f32(
                  false, a, false, b, (short)0, acc, false, false);
    }
    // C layout: VGPR r -> M = half*8 + r, N = lane&15
    float* crow = Cm + (size_t)row0 * ncols + col0 + m;
#pragma unroll
    for (int r = 0; r < 8; ++r)
        crow[(size_t)(half * 8 + r) * ncols] = acc[r];
}

// ---------- attention logits: one thread per (node, head) ----------
__global__ void att_logits_kernel(const float* __restrict__ h,
                                  const float* __restrict__ asrc,
                                  const float* __restrict__ adst,
                                  float* __restrict__ als, float* __restrict__ ald,
                                  int n) {
    int i = blockIdx.x * blockDim.x + threadIdx.x;
    if (i >= n * 2) return;
    int node = i >> 1, hd = i & 1;
    const float* hp = h + (size_t)node * HC + hd * C_HID;
    const float* as = asrc + hd * C_HID;
    const float* ad = adst + hd * C_HID;
    float ss = 0.f, sd = 0.f;
#pragma unroll 4
    for (int c = 0; c < C_HID; ++c) {
        float v = hp[c];
        ss += v * as[c];
        sd += v * ad[c];
    }
    als[i] = ss;
    ald[i] = sd;
}

// ---------- edge pass A: segment max (monotonic-uint atomicMax) ----------
__global__ void edge_max_kernel(const int* __restrict__ ei,
                                const float* __restrict__ als,
                                const float* __restrict__ ald,
                                unsigned* __restrict__ mbuf, int E, int ET) {
    int e = blockIdx.x * blockDim.x + threadIdx.x;
    if (e >= ET) return;
    int s, d; edge_sd(ei, e, E, s, d);
#pragma unroll
    for (int h = 0; h < 2; ++h) {
        float v = als[s * 2 + h] + ald[d * 2 + h];
        v = v > 0.f ? v : NEG_SLOPE * v;
        atomicMax(&mbuf[d * 2 + h], f2mono(v));
    }
}

// ---------- edge pass B: segment sum of exp(e - max) ----------
__global__ void edge_sum_kernel(const int* __restrict__ ei,
                                const float* __restrict__ als,
                                const float* __restrict__ ald,
                                const unsigned* __restrict__ mbuf,
                                float* __restrict__ sbuf, int E, int ET) {
    int e = blockIdx.x * blockDim.x + threadIdx.x;
    if (e >= ET) return;
    int s, d; edge_sd(ei, e, E, s, d);
#pragma unroll
    for (int h = 0; h < 2; ++h) {
        float v = als[s * 2 + h] + ald[d * 2 + h];
        v = v > 0.f ? v : NEG_SLOPE * v;
        atomicAdd(&sbuf[d * 2 + h], __expf(v - mono2f(mbuf[d * 2 + h])));
    }
}

// ---------- edge pass C: wave per edge, lane = 4 channels, float4 gather ----
__global__ void edge_agg_kernel(const int* __restrict__ ei,
                                const float* __restrict__ als,
                                const float* __restrict__ ald,
                                const unsigned* __restrict__ mbuf,
                                const float* __restrict__ sbuf,
                                const float* __restrict__ h,
                                float* __restrict__ out, int E, int ET) {
    int wid  = (int)(((long long)blockIdx.x * blockDim.x + threadIdx.x) >> 5);
    int lane = threadIdx.x & 31;
    if (wid >= ET) return;
    int s, d; edge_sd(ei, wid, E, s, d);
    int hd = lane >> 4;                    // channel = lane*4 .. +3 ; head = ch>>6
    float v = als[s * 2 + hd] + ald[d * 2 + hd];
    v = v > 0.f ? v : NEG_SLOPE * v;
    float alpha = __expf(v - mono2f(mbuf[d * 2 + hd])) / sbuf[d * 2 + hd];
    const float4 val = ((const float4*)(h + (size_t)s * HC))[lane];
    float* ob = out + (size_t)d * HC + lane * 4;
    atomicAdd(ob + 0, val.x * alpha);
    atomicAdd(ob + 1, val.y * alpha);
    atomicAdd(ob + 2, val.z * alpha);
    atomicAdd(ob + 3, val.w * alpha);
}

// ---------- mean over heads: [N,2,64] -> [N,64] ----------
__global__ void head_mean_kernel(const float* __restrict__ agg,
                                 float* __restrict__ x3, int n) {
    int i = blockIdx.x * blockDim.x + threadIdx.x;
    if (i >= n * C_HID) return;
    int node = i >> 6, c = i & 63;
    const float* a = agg + (size_t)node * HC;
    x3[i] = 0.5f * (a[c] + a[C_HID + c]);
}

// ---------- final linear: [N,64] @ [64,40] + bf ----------
__global__ void final_linear_kernel(const float* __restrict__ x,
                                    const float* __restrict__ Wf,
                                    const float* __restrict__ bf,
                                    float* __restrict__ out, int n) {
    int i = blockIdx.x * blockDim.x + threadIdx.x;
    if (i >= n * OUT_CH) return;
    int node = i / OUT_CH, o = i % OUT_CH;
    const float* xr = x + (size_t)node * C_HID;
    float acc = bf[o];
#pragma unroll 8
    for (int c = 0; c < C_HID; ++c)
        acc += xr[c] * Wf[c * OUT_CH + o];
    out[i] = acc;
}

extern "C" void kernel_launch(void* const* d_in, const int* in_sizes, int n_in,
                              void* d_out, int out_size, void* d_ws, size_t ws_size,
                              hipStream_t stream) {
    const float* x      = (const float*)d_in[0];
    const int*   ei     = (const int*)d_in[1];
    const float* bn0_g  = (const float*)d_in[2];
    const float* bn0_b  = (const float*)d_in[3];
    const float* W1     = (const float*)d_in[4];
    const float* a1_src = (const float*)d_in[5];
    const float* a1_dst = (const float*)d_in[6];
    // d_in[7] = b1 (cancels inside BN1)
    const float* bn1_g  = (const float*)d_in[8];
    const float* bn1_b  = (const float*)d_in[9];
    const float* W2     = (const float*)d_in[10];
    const float* a2_src = (const float*)d_in[11];
    const float* a2_dst = (const float*)d_in[12];
    // d_in[13] = b2 (cancels inside BN2)
    const float* bn2_g  = (const float*)d_in[14];
    const float* bn2_b  = (const float*)d_in[15];
    const float* Wf     = (const float*)d_in[16];
    const float* bf     = (const float*)d_in[17];
    float* out = (float*)d_out;

    const int N = N_NODES, E = N_EDGES, ET = N_EDGES + N_NODES;

    // workspace carve-up
    char* w = (char*)d_ws;
    float*    A0   = (float*)w;                 w += (size_t)N * HC * 4;   // BN'd features
    float*    P    = (float*)w;                 w += (size_t)N * HC * 4;   // projection h
    float*    AG   = (float*)w;                 w += (size_t)N * HC * 4;   // aggregate out
    float*    X3   = (float*)w;                 w += (size_t)N * C_HID * 4;
    float*    ALS  = (float*)w;                 w += (size_t)N * 2 * 4;
    float*    ALD  = (float*)w;                 w += (size_t)N * 2 * 4;
    unsigned* MB   = (unsigned*)w;              w += (size_t)N * 2 * 4;
    float*    SB   = (float*)w;                 w += (size_t)N * 2 * 4;
    float*    MU   = (float*)w;                 w += 128 * 4;
    float*    VAR  = (float*)w;                 w += 128 * 4;

    dim3 blk256(256);
    dim3 gemm_blk(128);
    dim3 gemm_grid(N / 16, HC / 64);
    int g_elems128 = (N * HC + 255) / 256;
    int g_nodeheads = (N * 2 + 255) / 256;
    int g_edges = (ET + 255) / 256;
    int g_edge_waves = (ET * 32 + 255) / 256;

    // ---- BN0 ----
    bn_stats_kernel<<<IN_CH, blk256, 0, stream>>>(x, N, IN_CH, MU, VAR);
    bn_apply_kernel<<<g_elems128, blk256, 0, stream>>>(x, A0, MU, VAR, bn0_g, bn0_b,
                                                       (long long)N * IN_CH, IN_CH, 0);
    // ---- layer 1 ----
    gemm_wmma_f32_kernel<<<gemm_grid, gemm_blk, 0, stream>>>(A0, W1, P, N, IN_CH, HC);
    att_logits_kernel<<<g_nodeheads, blk256, 0, stream>>>(P, a1_src, a1_dst, ALS, ALD, N);
    hipMemsetAsync(MB, 0, (size_t)N * 2 * 4, stream);   // 0 < mono(any finite float)
    hipMemsetAsync(SB, 0, (size_t)N * 2 * 4, stream);
    hipMemsetAsync(AG, 0, (size_t)N * HC * 4, stream);
    edge_max_kernel<<<g_edges, blk256, 0, stream>>>(ei, ALS, ALD, MB, E, ET);
    edge_sum_kernel<<<g_edges, blk256, 0, stream>>>(ei, ALS, ALD, MB, SB, E, ET);
    edge_agg_kernel<<<g_edge_waves, blk256, 0, stream>>>(ei, ALS, ALD, MB, SB, P, AG, E, ET);
    // ---- BN1 + ELU ----
    bn_stats_kernel<<<HC, blk256, 0, stream>>>(AG, N, HC, MU, VAR);
    bn_apply_kernel<<<g_elems128, blk256, 0, stream>>>(AG, A0, MU, VAR, bn1_g, bn1_b,
                                                       (long long)N * HC, HC, 1);
    // ---- layer 2 ----
    gemm_wmma_f32_kernel<<<gemm_grid, gemm_blk, 0, stream>>>(A0, W2, P, N, HC, HC);
    att_logits_kernel<<<g_nodeheads, blk256, 0, stream>>>(P, a2_src, a2_dst, ALS, ALD, N);
    hipMemsetAsync(MB, 0, (size_t)N * 2 * 4, stream);
    hipMemsetAsync(SB, 0, (size_t)N * 2 * 4, stream);
    hipMemsetAsync(AG, 0, (size_t)N * HC * 4, stream);
    edge_max_kernel<<<g_edges, blk256, 0, stream>>>(ei, ALS, ALD, MB, E, ET);
    edge_sum_kernel<<<g_edges, blk256, 0, stream>>>(ei, ALS, ALD, MB, SB, E, ET);
    edge_agg_kernel<<<g_edge_waves, blk256, 0, stream>>>(ei, ALS, ALD, MB, SB, P, AG, E, ET);
    // ---- head mean + BN2 + ELU ----
    head_mean_kernel<<<(N * C_HID + 255) / 256, blk256, 0, stream>>>(AG, X3, N);
    bn_stats_kernel<<<C_HID, blk256, 0, stream>>>(X3, N, C_HID, MU, VAR);
    bn_apply_kernel<<<(N * C_HID + 255) / 256, blk256, 0, stream>>>(X3, X3, MU, VAR,
                                                       bn2_g, bn2_b,
                                                       (long long)N * C_HID, C_HID, 1);
    // ---- final linear ----
    final_linear_kernel<<<(N * OUT_CH + 255) / 256, blk256, 0, stream>>>(X3, Wf, bf, out, N);
    (void)in_sizes; (void)n_in; (void)out_size; (void)ws_size;
}